// VectorQuantizerEMA_54906861912275
// MI455X (gfx1250) — compile-verified
//
#include <hip/hip_runtime.h>
#include <hip/hip_bf16.h>
#include <math.h>

typedef __attribute__((ext_vector_type(2))) float v2f;
typedef __attribute__((ext_vector_type(4))) float v4f;
typedef __attribute__((ext_vector_type(8))) float v8f;

#define NROWS   65536      // B*H*W = 16*64*64
#define NEMB    1024
#define EDIM    64
#define HW      4096       // H*W
#define BSTRIDE 262144     // C*H*W = 64*4096

// -------- init: zero counts + loss accumulator ------------------------------
__global__ void vq_init(unsigned* __restrict__ counts, float* __restrict__ loss_acc) {
    int t = blockIdx.x * 256 + threadIdx.x;
    if (t < NEMB) counts[t] = 0u;
    if (t == 0) *loss_acc = 0.0f;
}

// -------- main kernel: fp32 WMMA distance GEMM + per-row argmin -------------
// Block = 256 threads = 8 waves; each wave owns a 16-row tile of flat x.
// LDS: embedding [1024][64] (256KB) + enorm [1024] (4KB) = 260KB of 320KB WGP LDS.
__launch_bounds__(256)
__global__ void vq_argmin(const float* __restrict__ x, const float* __restrict__ E,
                          int* __restrict__ idx_out, unsigned* __restrict__ counts,
                          float* __restrict__ loss_acc) {
    extern __shared__ float smem[];
    float* sE = smem;              // 65536 floats
    float* sN = smem + NEMB * EDIM; // 1024 floats

    const int tid = threadIdx.x;

    // cooperative stage of codebook into LDS (128b vectorized)
    {
        const v4f* Ev = (const v4f*)E;
        v4f* sEv = (v4f*)sE;
        for (int i = tid; i < NEMB * EDIM / 4; i += 256) sEv[i] = Ev[i];
    }
    __syncthreads();
    // codebook row norms
    for (int k = tid; k < NEMB; k += 256) {
        float s = 0.0f;
        const float* r = sE + k * EDIM;
        #pragma unroll 8
        for (int d = 0; d < EDIM; ++d) { float v = r[d]; s = fmaf(v, v, s); }
        sN[k] = s;
    }
    __syncthreads();

    const int wave = tid >> 5;
    const int lane = tid & 31;
    const int m    = lane & 15;         // row/col within 16-wide tile
    const int koff = (lane >> 4) * 2;   // f32 WMMA: lanes 16-31 hold K+2,K+3
    const int rowbase = (blockIdx.x * 8 + wave) * 16;
    const int row = rowbase + m;

    // A fragments: a[kk][v] = x_flat[row][4*kk + koff + v]
    // x is BCHW: x_flat[n][d] = in[(n>>12)*BSTRIDE + d*HW + (n & (HW-1))]
    const float* xrow = x + (size_t)(row >> 12) * BSTRIDE + (row & (HW - 1));
    v2f a[16];
    float sq = 0.0f;  // partial sum of squares of this lane's held x elements
    #pragma unroll
    for (int kk = 0; kk < 16; ++kk) {
        float a0 = xrow[(size_t)(4 * kk + koff) * HW];
        float a1 = xrow[(size_t)(4 * kk + koff + 1) * HW];
        a[kk] = (v2f){a0, a1};
        sq = fmaf(a0, a0, sq);
        sq = fmaf(a1, a1, sq);
    }
    // full-wave sum -> total of all 16 row norms for this wave's tile
    #pragma unroll
    for (int msk = 16; msk >= 1; msk >>= 1) sq += __shfl_xor(sq, msk, 32);

    float minv[8];
    int   mini[8];
    #pragma unroll
    for (int j = 0; j < 8; ++j) { minv[j] = 3.4e38f; mini[j] = 0; }

    // sweep 64 column tiles of 16 codebook rows each
    for (int ct = 0; ct < 64; ++ct) {
        v8f acc = {};
        const float* bptr = sE + (size_t)(ct * 16 + m) * EDIM + koff;
        #pragma unroll
        for (int kk = 0; kk < 16; ++kk) {
            v2f bf = *(const v2f*)(bptr + 4 * kk);  // 8B-aligned LDS read
            acc = __builtin_amdgcn_wmma_f32_16x16x4_f32(
                false, a[kk], false, bf, (short)0, acc, false, false);
        }
        const float en = sN[ct * 16 + m];
        const int   c  = ct * 16 + m;
        #pragma unroll
        for (int j = 0; j < 8; ++j) {
            float t = fmaf(-2.0f, acc[j], en);   // ||e||^2 - 2 x.e
            if (t < minv[j]) { minv[j] = t; mini[j] = c; }  // strict <: first-min
        }
    }

    // argmin across the 16-lane half that shares each row (xor bits 0..3)
    #pragma unroll
    for (int msk = 8; msk >= 1; msk >>= 1) {
        #pragma unroll
        for (int j = 0; j < 8; ++j) {
            float ov = __shfl_xor(minv[j], msk, 32);
            int   oi = __shfl_xor(mini[j], msk, 32);
            if (ov < minv[j] || (ov == minv[j] && oi < mini[j])) {
                minv[j] = ov; mini[j] = oi;
            }
        }
    }

    // lane 0 owns rows rowbase+0..7, lane 16 owns rows rowbase+8..15
    if ((lane & 15) == 0) {
        float s = (lane == 0) ? sq : 0.0f;  // add tile's total ||x||^2 once
        const int rb = rowbase + (lane >> 4) * 8;
        #pragma unroll
        for (int j = 0; j < 8; ++j) {
            s += minv[j];
            idx_out[rb + j] = mini[j];
            atomicAdd(&counts[mini[j]], 1u);
        }
        atomicAdd(loss_acc, s);
    }
}

// -------- finalize: loss scalar + perplexity --------------------------------
__global__ void vq_final(const unsigned* __restrict__ counts,
                         const float* __restrict__ loss_acc,
                         float* __restrict__ out_loss, float* __restrict__ out_perp) {
    __shared__ float red[256];
    float s = 0.0f;
    for (int k = threadIdx.x; k < NEMB; k += 256) {
        float p = (float)counts[k] * (1.0f / (float)NROWS);
        s -= p * logf(p + 1e-10f);
    }
    red[threadIdx.x] = s;
    __syncthreads();
    for (int st = 128; st > 0; st >>= 1) {
        if (threadIdx.x < st) red[threadIdx.x] += red[threadIdx.x + st];
        __syncthreads();
    }
    if (threadIdx.x == 0) {
        *out_perp = expf(red[0]);
        *out_loss = 0.25f * (*loss_acc) * (1.0f / ((float)NROWS * (float)EDIM));
    }
}

// -------- q_ste: gather codebook rows into BDHW layout ----------------------
__global__ void vq_qste(const int* __restrict__ idx, const float* __restrict__ E,
                        float* __restrict__ out) {
    size_t o = (size_t)blockIdx.x * 256 + threadIdx.x;   // linear over B,D,H,W
    int w = (int)(o & 63);
    int h = (int)((o >> 6) & 63);
    int d = (int)((o >> 12) & 63);
    int b = (int)(o >> 18);
    int n = (b << 12) | (h << 6) | w;
    out[o] = E[(size_t)idx[n] * EDIM + d];
}

// -------- encodings: one-hot (N,1024), streamed with NT stores --------------
__global__ void vq_encodings(const int* __restrict__ idx, float* __restrict__ enc) {
    int rowq = blockIdx.x;                 // one block per flat row
    int k = idx[rowq];
    v4f* out = (v4f*)(enc + (size_t)rowq * NEMB);
    int c0 = threadIdx.x * 4;
    v4f v = (v4f){0.0f, 0.0f, 0.0f, 0.0f};
    if (k >= c0 && k < c0 + 4) v[k - c0] = 1.0f;
    __builtin_nontemporal_store(v, &out[threadIdx.x]);
}

// -------- indices as float --------------------------------------------------
__global__ void vq_indices(const int* __restrict__ idx, float* __restrict__ out) {
    int n = blockIdx.x * 256 + threadIdx.x;
    out[n] = (float)idx[n];
}

extern "C" void kernel_launch(void* const* d_in, const int* in_sizes, int n_in,
                              void* d_out, int out_size, void* d_ws, size_t ws_size,
                              hipStream_t stream) {
    const float* x = (const float*)d_in[0];   // (16,64,64,64) BCHW
    const float* E = (const float*)d_in[1];   // (1024,64)

    float* out      = (float*)d_out;
    float* out_loss = out;                                     // [1]
    float* out_q    = out + 1;                                 // [4194304]
    float* out_perp = out + 1 + 4194304;                       // [1]
    float* out_enc  = out + 2 + 4194304;                       // [67108864]
    float* out_idx  = out + 2 + 4194304 + 67108864;            // [65536]

    char* ws = (char*)d_ws;
    float*    loss_acc = (float*)ws;                 // @0
    unsigned* counts   = (unsigned*)(ws + 256);      // @256  (1024 u32)
    int*      idxbuf   = (int*)(ws + 256 + 4096);    // @4352 (65536 i32)

    const size_t smem = (size_t)(NEMB * EDIM + NEMB) * sizeof(float); // 266,240 B

    vq_init     <<<4,     256, 0,    stream>>>(counts, loss_acc);
    vq_argmin   <<<512,   256, smem, stream>>>(x, E, idxbuf, counts, loss_acc);
    vq_final    <<<1,     256, 0,    stream>>>(counts, loss_acc, out_loss, out_perp);
    vq_qste     <<<16384, 256, 0,    stream>>>(idxbuf, E, out_q);
    vq_encodings<<<65536, 256, 0,    stream>>>(idxbuf, out_enc);
    vq_indices  <<<256,   256, 0,    stream>>>(idxbuf, out_idx);
}